// MemoryEfficientCrossAttention_tri_55774445305906
// MI455X (gfx1250) — compile-verified
//
#include <hip/hip_runtime.h>

// ---------------------------------------------------------------------------
// MemoryEfficientCrossAttention (triplane axial attention) for gfx1250
//   stage 1: q = x @ Wq^T                 (fp32 WMMA 16x16x4, 16x64 strip/wave)
//   stage 2: gathered 128-context attention per (batch,loc,head)  (VALU)
//   stage 3: out = attn_out @ Wo^T + bo   (fp32 WMMA 16x16x4, 16x64 strip/wave)
// ---------------------------------------------------------------------------

typedef float v2f __attribute__((ext_vector_type(2)));
typedef float v8f __attribute__((ext_vector_type(8)));

#define LDIM 1024   // 32*32 locations per plane
#define QD   512    // feature dim (= HEADS*DIM_HEAD)
#define NH   8
#define DH   64

// ---------------------------------------------------------------------------
// GEMM: C[n,m] = sum_k X[n,k] * W[m,k] (+ bias[m])
// Each wave computes a 16x64 strip = 4 C tiles sharing one A fragment per
// k-step: 1 A load + 4 B loads + 4 WMMAs  (1.25 loads per wmma vs 2.0 for a
// single-tile wave).  A 16x4 layout (ISA 7.12.2): lane&15 = M row, lane>>4
// selects K pair {0,1}/{2,3}; fragment = consecutive K pair -> one b64 load.
// B 4x16 mirrors with lane&15 = N column.
// ---------------------------------------------------------------------------
__global__ __launch_bounds__(256) void gemm_xwt_f32(const float* __restrict__ X,
                                                    const float* __restrict__ W,
                                                    const float* __restrict__ bias,
                                                    float* __restrict__ C,
                                                    int nrows)
{
    const int lane = threadIdx.x & 31;
    const int wave = threadIdx.x >> 5;
    const int job  = blockIdx.x * 8 + wave;     // 16x64 strip id
    const int row_tiles = nrows >> 4;           // 192
    const int n0 = (job % row_tiles) << 4;      // output row tile (x rows)
    const int m0 = (job / row_tiles) << 6;      // output 64-wide column strip

    const int r  = lane & 15;                   // A row / B column within tile
    const int kp = (lane >> 4) << 1;            // K-pair offset: 0 or 2

    const float* aptr = X + (size_t)(n0 + r) * QD + kp;
    const float* bptr = W + (size_t)(m0 + r) * QD + kp;

    v8f acc0 = {}, acc1 = {}, acc2 = {}, acc3 = {};
#pragma unroll 4
    for (int k = 0; k < QD; k += 4) {
        v2f a  = *(const v2f*)(aptr + k);
        v2f b0 = *(const v2f*)(bptr + k);
        v2f b1 = *(const v2f*)(bptr + 16 * QD + k);
        v2f b2 = *(const v2f*)(bptr + 32 * QD + k);
        v2f b3 = *(const v2f*)(bptr + 48 * QD + k);
        acc0 = __builtin_amdgcn_wmma_f32_16x16x4_f32(false, a, false, b0,
                                                     (short)0, acc0, false, false);
        acc1 = __builtin_amdgcn_wmma_f32_16x16x4_f32(false, a, false, b1,
                                                     (short)0, acc1, false, false);
        acc2 = __builtin_amdgcn_wmma_f32_16x16x4_f32(false, a, false, b2,
                                                     (short)0, acc2, false, false);
        acc3 = __builtin_amdgcn_wmma_f32_16x16x4_f32(false, a, false, b3,
                                                     (short)0, acc3, false, false);
    }

    // C layout (ISA 7.12.2): VGPR rr -> M = rr (lanes 0-15) / rr+8 (lanes 16-31),
    // N = lane&15.
    const int mrow = (lane >> 4) << 3;
    v8f accs[4] = {acc0, acc1, acc2, acc3};
#pragma unroll
    for (int t = 0; t < 4; ++t) {
        const int   mc = m0 + t * 16 + r;
        const float bv = bias ? bias[mc] : 0.0f;
        float* cp = C + (size_t)(n0 + mrow) * QD + mc;
#pragma unroll
        for (int rr = 0; rr < 8; ++rr)
            cp[(size_t)rr * QD] = accs[t][rr] + bv;
    }
}

// ---------------------------------------------------------------------------
// Context gather map: for query plane `batch` at (row j, col i), context c in
// [0,128) maps to source plane `src` and flat location `loc` (row*32+col).
// Derived from the reference's concat/broadcast pattern (mid = h/2-1 = 15).
// ---------------------------------------------------------------------------
__device__ __forceinline__ void ctx_map(int batch, int i, int j, int c,
                                        int& src, int& loc)
{
    const int mid = 15;
    const int cc  = c & 31;
    if (batch == 0) {                 // queries from plane1
        if      (c <  32) { src = 1; loc = cc  * 32 + i;        } // col i of f2
        else if (c <  64) { src = 1; loc = mid * 32 + cc;       } // row mid of f2
        else if (c <  96) { src = 2; loc = cc  * 32 + mid;      } // col mid of f3
        else              { src = 2; loc = j   * 32 + cc;       } // row j of f3
    } else if (batch == 1) {          // queries from plane2
        if      (c <  32) { src = 0; loc = cc  * 32 + i;        } // col i of f1
        else if (c <  64) { src = 0; loc = mid * 32 + cc;       } // row mid of f1
        else if (c <  96) { src = 2; loc = cc  * 32 + (31 - j); } // col 31-j of f3
        else              { src = 2; loc = mid * 32 + cc;       } // row mid of f3
    } else {                          // queries from plane3
        if      (c <  32) { src = 0; loc = cc  * 32 + mid;      } // col mid of f1
        else if (c <  64) { src = 0; loc = j   * 32 + cc;       } // row j of f1
        else if (c <  96) { src = 2; loc = cc  * 32 + mid;      } // col mid of f3 (ref uses f3 here)
        else              { src = 2; loc = (31 - i) * 32 + cc;  } // row 31-i of f3
    }
}

// ---------------------------------------------------------------------------
// Attention: grid = 3*1024 blocks (one per batch,location), 256 thr = 8 waves,
// wave w = head w.  Each lane owns 2 of the 64 d-components.  K == V.
// ---------------------------------------------------------------------------
__global__ __launch_bounds__(256) void attn_kernel(const float* __restrict__ q,
                                                   float* __restrict__ out)
{
    const int bl    = blockIdx.x;          // 0..3071
    const int batch = bl >> 10;
    const int l     = bl & 1023;
    const int i     = l & 31;              // column
    const int j     = l >> 5;              // row
    const int head  = threadIdx.x >> 5;
    const int lane  = threadIdx.x & 31;

    const float* qv = q + ((size_t)(batch * LDIM + l)) * QD + head * DH;
    const v2f qf = *(const v2f*)(qv + lane * 2);

    // ---- pass 1: scores (dot per context, wave32 butterfly reduce) ----
    float s[4];
    for (int c = 0; c < 128; ++c) {
        int src, loc;
        ctx_map(batch, i, j, c, src, loc);
        const float* kv = q + ((size_t)(src * LDIM + loc)) * QD + head * DH;
        const v2f kf = *(const v2f*)(kv + lane * 2);
        float part = qf.x * kf.x + qf.y * kf.y;
#pragma unroll
        for (int off = 16; off; off >>= 1)
            part += __shfl_xor(part, off, 32);
        if (lane == (c & 31)) s[c >> 5] = part * 0.125f;   // 1/sqrt(64)
    }

    // ---- softmax over 128 (4 regs x 32 lanes) ----
    float mx = fmaxf(fmaxf(s[0], s[1]), fmaxf(s[2], s[3]));
#pragma unroll
    for (int off = 16; off; off >>= 1)
        mx = fmaxf(mx, __shfl_xor(mx, off, 32));
    float sum = 0.0f;
#pragma unroll
    for (int t = 0; t < 4; ++t) { s[t] = __expf(s[t] - mx); sum += s[t]; }
#pragma unroll
    for (int off = 16; off; off >>= 1)
        sum += __shfl_xor(sum, off, 32);
    const float inv = 1.0f / sum;

    // ---- pass 2: out = P @ V (V == K), contexts re-gathered (L2 hot) ----
    float ax = 0.0f, ay = 0.0f;
    for (int c = 0; c < 128; ++c) {
        int src, loc;
        ctx_map(batch, i, j, c, src, loc);
        const float* kv = q + ((size_t)(src * LDIM + loc)) * QD + head * DH;
        const v2f kf = *(const v2f*)(kv + lane * 2);
        const float p = __shfl(s[c >> 5], c & 31, 32);
        ax += p * kf.x;
        ay += p * kf.y;
    }

    float* op = out + ((size_t)(batch * LDIM + l)) * QD + head * DH + lane * 2;
    op[0] = ax * inv;
    op[1] = ay * inv;
}

// ---------------------------------------------------------------------------
extern "C" void kernel_launch(void* const* d_in, const int* in_sizes, int n_in,
                              void* d_out, int out_size, void* d_ws, size_t ws_size,
                              hipStream_t stream)
{
    const float* x  = (const float*)d_in[0];   // (3,1024,512)
    const float* Wq = (const float*)d_in[1];   // (512,512) row-major (inner,qd)
    const float* Wo = (const float*)d_in[2];   // (512,512) row-major (qd,inner)
    const float* bo = (const float*)d_in[3];   // (512,)

    float* qbuf = (float*)d_ws;                // 3*1024*512 floats (6 MB)
    float* abuf = qbuf + 3 * LDIM * QD;        // 3*1024*512 floats (6 MB)
    float* outp = (float*)d_out;               // (3,1024,512)

    const int nrows  = 3 * LDIM;                       // 3072
    const int jobs   = (nrows / 16) * (QD / 64);       // 1536 strips
    const int blocks = jobs / 8;                       // 8 waves (strips) / block

    // stage 1: q = x @ Wq^T
    gemm_xwt_f32<<<blocks, 256, 0, stream>>>(x, Wq, nullptr, qbuf, nrows);
    // stage 2: gathered attention (K == V)
    attn_kernel<<<3 * LDIM, 256, 0, stream>>>(qbuf, abuf);
    // stage 3: out = attn_out @ Wo^T + bo
    gemm_xwt_f32<<<blocks, 256, 0, stream>>>(abuf, Wo, bo, outp, nrows);
}